// CatAttention_26895085208135
// MI455X (gfx1250) — compile-verified
//
#include <hip/hip_runtime.h>
#include <hip/hip_bf16.h>
#include <math.h>

#define Bn 32
#define Sn 4096
#define Dn 512
#define Hn 512

typedef __attribute__((ext_vector_type(2))) float v2f;
typedef __attribute__((ext_vector_type(8))) float v8f;

// Branch-free tanh: keeps EXEC all-ones (WMMA requirement) and avoids the
// libm slow-path saveexec region. Clamp prevents inf/inf NaN; tanh saturates
// to +/-1 well before |x|=15 in f32 anyway.
__device__ __forceinline__ float fast_tanh(float x) {
    float xc = fminf(fmaxf(x, -15.f), 15.f);
    float t  = __expf(2.f * xc);
    return __fdividef(t - 1.f, t + 1.f);
}

// ---------------------------------------------------------------------------
// K1: tp[b,h] = dot(targ[b,:], W[h, 0:512]);  also zero ctx accumulator.
// One thread per (b,h): B*H = 16384 threads.
// ---------------------------------------------------------------------------
__global__ void k1_targproj(const float* __restrict__ targ,
                            const float* __restrict__ W,
                            float* __restrict__ tp,
                            float* __restrict__ ctx_zero) {
    int i = blockIdx.x * 256 + threadIdx.x;     // i < B*H
    int b = i >> 9;                             // /512
    int h = i & (Hn - 1);
    const float* tr = targ + (size_t)b * Dn;
    const float* wr = W + (size_t)h * (2 * Dn); // row h, cols 0..511 = Wt
    float acc = 0.f;
#pragma unroll 8
    for (int d = 0; d < Dn; ++d) acc += tr[d] * wr[d];
    tp[i] = acc;
    ctx_zero[i] = 0.f;                          // B*D == B*H elements
}

// ---------------------------------------------------------------------------
// K2: main WMMA kernel.
//   e[b,s] = sum_h V[h] * tanh( tp[b,h] + dot(src[b,s,:], W[h,512:1024]) )
// Block: 256 threads = 8 waves. Each wave owns TWO 16-row s-tiles (M=s) and
// sweeps h-tiles (N=h) in PAIRS: 2x2 register blocking -> 4 v8f accumulators,
// 4 global_load_b64 per 4 v_wmma_f32_16x16x4_f32 (1 load/WMMA).
// Grid: (S/256, B).
// ---------------------------------------------------------------------------
__global__ void __launch_bounds__(256)
k2_energy(const float* __restrict__ src,
          const float* __restrict__ W,
          const float* __restrict__ V,
          const float* __restrict__ tp,
          float* __restrict__ e_ws) {
    const int lane = threadIdx.x & 31;
    const int wv   = threadIdx.x >> 5;
    const int b    = blockIdx.y;
    const int s0   = blockIdx.x * 256 + wv * 32;   // this wave: rows s0..s0+31

    const int mrow = lane & 15;          // A: s-row in tile / B: h-row in tile
    const int ko   = (lane >> 4) << 1;   // lanes 16-31 carry K=2,3

    // A operand bases: src[b][s0+mrow][*] and src[b][s0+16+mrow][*]
    const float* aPtr0 = src + ((size_t)b * Sn + (s0 + mrow)) * Dn + ko;
    const float* aPtr1 = aPtr0 + (size_t)16 * Dn;

    float eacc0[8], eacc1[8];
#pragma unroll
    for (int r = 0; r < 8; ++r) { eacc0[r] = 0.f; eacc1[r] = 0.f; }

    const float* tpb = tp + (size_t)b * Hn;

    for (int ht = 0; ht < Hn / 16; ht += 2) {
        const int h0 = ht * 16;
        // B operand bases: Ws[h][*] = W[h][512 + *]
        const float* bPtr0 = W + (size_t)(h0 + mrow) * (2 * Dn) + Dn + ko;
        const float* bPtr1 = bPtr0 + (size_t)16 * (2 * Dn);

        v8f c00 = {}, c01 = {}, c10 = {}, c11 = {};
#pragma unroll 4
        for (int k = 0; k < Dn; k += 4) {
            v2f a0 = *(const v2f*)(aPtr0 + k);
            v2f a1 = *(const v2f*)(aPtr1 + k);
            v2f b0 = *(const v2f*)(bPtr0 + k);
            v2f b1 = *(const v2f*)(bPtr1 + k);
            c00 = __builtin_amdgcn_wmma_f32_16x16x4_f32(
                      false, a0, false, b0, (short)0, c00, false, false);
            c01 = __builtin_amdgcn_wmma_f32_16x16x4_f32(
                      false, a0, false, b1, (short)0, c01, false, false);
            c10 = __builtin_amdgcn_wmma_f32_16x16x4_f32(
                      false, a1, false, b0, (short)0, c10, false, false);
            c11 = __builtin_amdgcn_wmma_f32_16x16x4_f32(
                      false, a1, false, b1, (short)0, c11, false, false);
        }

        // Epilogue: lane's N column is h0 + (lane&15) (tile 0) / +16 (tile 1).
        const float tp0 = tpb[h0 + mrow];
        const float tp1 = tpb[h0 + 16 + mrow];
        const float v0  = V[h0 + mrow];
        const float v1  = V[h0 + 16 + mrow];
#pragma unroll
        for (int r = 0; r < 8; ++r) {
            eacc0[r] += v0 * fast_tanh(c00[r] + tp0)
                      + v1 * fast_tanh(c01[r] + tp1);
            eacc1[r] += v0 * fast_tanh(c10[r] + tp0)
                      + v1 * fast_tanh(c11[r] + tp1);
        }
    }

    // Reduce over the 16-lane N-groups (lanes 0-15 hold M=r, 16-31 hold M=r+8).
#pragma unroll
    for (int r = 0; r < 8; ++r) {
        float x = eacc0[r];
        x += __shfl_xor(x, 8, 16);
        x += __shfl_xor(x, 4, 16);
        x += __shfl_xor(x, 2, 16);
        x += __shfl_xor(x, 1, 16);
        eacc0[r] = x;
        float y = eacc1[r];
        y += __shfl_xor(y, 8, 16);
        y += __shfl_xor(y, 4, 16);
        y += __shfl_xor(y, 2, 16);
        y += __shfl_xor(y, 1, 16);
        eacc1[r] = y;
    }
    if ((lane & 15) == 0) {
        const int mbase = (lane >> 4) * 8;   // lane 0 -> rows 0..7, lane 16 -> 8..15
        float* eb0 = e_ws + (size_t)b * Sn + s0 + mbase;
        float* eb1 = eb0 + 16;
#pragma unroll
        for (int r = 0; r < 8; ++r) { eb0[r] = eacc0[r]; eb1[r] = eacc1[r]; }
    }
}

// ---------------------------------------------------------------------------
// K3: per-b softmax over S, times mask -> att (first section of d_out).
// One block per b.
// ---------------------------------------------------------------------------
__global__ void __launch_bounds__(256)
k3_softmax(const float* __restrict__ e_ws,
           const float* __restrict__ mask,
           float* __restrict__ att) {
    const int b = blockIdx.x;
    const int t = threadIdx.x;
    __shared__ float red[256];
    const float* eb = e_ws + (size_t)b * Sn;

    float m = -INFINITY;
    for (int s = t; s < Sn; s += 256) m = fmaxf(m, eb[s]);
    red[t] = m;
    __syncthreads();
    for (int off = 128; off > 0; off >>= 1) {
        if (t < off) red[t] = fmaxf(red[t], red[t + off]);
        __syncthreads();
    }
    const float mx = red[0];
    __syncthreads();

    float sum = 0.f;
    for (int s = t; s < Sn; s += 256) sum += __expf(eb[s] - mx);
    red[t] = sum;
    __syncthreads();
    for (int off = 128; off > 0; off >>= 1) {
        if (t < off) red[t] += red[t + off];
        __syncthreads();
    }
    const float inv = 1.0f / red[0];

    const float* mb = mask + (size_t)b * Sn;
    float* ab = att + (size_t)b * Sn;
    for (int s = t; s < Sn; s += 256)
        ab[s] = __expf(eb[s] - mx) * inv * mb[s];
}

// ---------------------------------------------------------------------------
// K4: ctx[b,d] += sum_{s in chunk} att[b,s] * src[b,s,d]
// Grid (8, B): 512-row s-chunks, att staged in LDS, float2 coalesced src,
// global_atomic_add_f32 accumulation into workspace.
// ---------------------------------------------------------------------------
__global__ void __launch_bounds__(256)
k4_ctx(const float* __restrict__ att,
       const float* __restrict__ src,
       float* __restrict__ ctx) {
    const int b  = blockIdx.y;
    const int sb = blockIdx.x * 512;
    const int t  = threadIdx.x;

    __shared__ float aw[512];
    for (int i = t; i < 512; i += 256) aw[i] = att[(size_t)b * Sn + sb + i];
    __syncthreads();

    const int d0 = t * 2;
    float ax = 0.f, ay = 0.f;
    const float* sp = src + ((size_t)b * Sn + sb) * Dn + d0;
#pragma unroll 4
    for (int s = 0; s < 512; ++s) {
        const float a = aw[s];
        const float2 v = *(const float2*)sp;
        sp += Dn;
        ax += a * v.x;
        ay += a * v.y;
    }
    atomicAdd(&ctx[(size_t)b * Dn + d0],     ax);
    atomicAdd(&ctx[(size_t)b * Dn + d0 + 1], ay);
}

// ---------------------------------------------------------------------------
// K5: hidden_output = ctx + targ  (second section of d_out)
// ---------------------------------------------------------------------------
__global__ void k5_add(const float* __restrict__ ctx,
                       const float* __restrict__ targ,
                       float* __restrict__ outh) {
    int i = blockIdx.x * 256 + threadIdx.x;   // < B*D
    outh[i] = ctx[i] + targ[i];
}

// ---------------------------------------------------------------------------
extern "C" void kernel_launch(void* const* d_in, const int* in_sizes, int n_in,
                              void* d_out, int out_size, void* d_ws, size_t ws_size,
                              hipStream_t stream) {
    const float* targ = (const float*)d_in[0];   // (B, D)
    const float* src  = (const float*)d_in[1];   // (B, S, D)
    const float* mask = (const float*)d_in[2];   // (B, S)
    const float* W    = (const float*)d_in[3];   // (D, 2D)
    const float* V    = (const float*)d_in[4];   // (D,)

    float* att  = (float*)d_out;                 // (B, S)
    float* outh = att + (size_t)Bn * Sn;         // (B, D)

    float* ws  = (float*)d_ws;
    float* tp  = ws;                              // B*H
    float* e   = tp + (size_t)Bn * Hn;            // B*S
    float* ctx = e + (size_t)Bn * Sn;             // B*D

    (void)in_sizes; (void)n_in; (void)out_size; (void)ws_size;

    k1_targproj<<<dim3((Bn * Hn) / 256), dim3(256), 0, stream>>>(targ, W, tp, ctx);
    k2_energy<<<dim3(Sn / 256, Bn), dim3(256), 0, stream>>>(src, W, V, tp, e);
    k3_softmax<<<dim3(Bn), dim3(256), 0, stream>>>(e, mask, att);
    k4_ctx<<<dim3(Sn / 512, Bn), dim3(256), 0, stream>>>(att, src, ctx);
    k5_add<<<dim3((Bn * Dn) / 256), dim3(256), 0, stream>>>(ctx, targ, outh);
}